// ShiTomasiAngleSparseBAD_66168266162568
// MI455X (gfx1250) — compile-verified
//
#include <hip/hip_runtime.h>
#include <hip/hip_bf16.h>
#include <math.h>

typedef float v2f __attribute__((ext_vector_type(2)));
typedef float v8f __attribute__((ext_vector_type(8)));

// ---------------------------------------------------------------------------
// Kernel 1: row-wise inclusive prefix sum via WMMA.
// out_tile = T(16x16) x U(16x16 upper-triangular ones), chained as 4 WMMAs of
// K=4. Inter-tile carry rides in the WMMA C accumulator; updated by shfl from
// lanes 15/31 (last column of the D tile). One wave per 16-row band.
// Loads are branch-free (clamped address + 0/1 mask) and hoisted ahead of the
// WMMA chain so the matrix ops issue back-to-back.
// ---------------------------------------------------------------------------
__global__ __launch_bounds__(32) void row_scan_wmma(
    const float* __restrict__ img, float* __restrict__ sat,
    int Hc, int Wc, int bandsPerImage)
{
    int blk  = blockIdx.x;
    int b    = blk / bandsPerImage;
    int rb   = blk % bandsPerImage;
    int row0 = rb * 16;
    int lane = threadIdx.x;
    int half = lane >> 4;      // 0: lanes 0-15, 1: lanes 16-31
    int m    = lane & 15;
    const size_t base = (size_t)b * Hc * Wc;

    // A-operand source row for this lane (clamped address, masked value)
    int   rowA  = row0 + m;
    int   rowAc = min(rowA, Hc - 1);
    float maskA = (rowA < Hc) ? 1.f : 0.f;
    const float* rowPtr = img + base + (size_t)rowAc * Wc + 2 * half;

    // B slices = U[4s:4s+4, :], U[k][n] = (k <= n); constants, hoisted.
    v2f bmat[4];
    #pragma unroll
    for (int s = 0; s < 4; ++s) {
        int kb = 4 * s + 2 * half;
        bmat[s] = v2f{ (kb     <= m) ? 1.f : 0.f,
                       (kb + 1 <= m) ? 1.f : 0.f };
    }

    bool fullBand = (row0 + 16 <= Hc);   // wave-uniform
    v8f carry = {0.f,0.f,0.f,0.f,0.f,0.f,0.f,0.f};
    int nColTiles = Wc / 16;             // W=1920 -> 120 full tiles
    for (int tx = 0; tx < nColTiles; ++tx) {
        // load all 4 A slices first (one clause, one wait)
        v2f a[4];
        #pragma unroll
        for (int s = 0; s < 4; ++s) {
            v2f av = *(const v2f*)(rowPtr + tx * 16 + 4 * s);  // 8B aligned
            a[s] = v2f{ av.x * maskA, av.y * maskA };
        }
        v8f c = carry;  // running row-prefix enters through the accumulator
        #pragma unroll
        for (int s = 0; s < 4; ++s)
            c = __builtin_amdgcn_wmma_f32_16x16x4_f32(
                    false, a[s], false, bmat[s], (short)0, c, false, false);

        // D layout: vgpr j, lane -> row (half*8 + j), col n = m
        float* outPtr = sat + base + tx * 16 + m;
        if (fullBand) {
            #pragma unroll
            for (int j = 0; j < 8; ++j)
                outPtr[(size_t)(row0 + half * 8 + j) * Wc] = c[j];
        } else {
            #pragma unroll
            for (int j = 0; j < 8; ++j) {
                int row = row0 + half * 8 + j;
                if (row < Hc) outPtr[(size_t)row * Wc] = c[j];
            }
        }
        // carry = prefix at n=15 (held by lane 15 / lane 31)
        int src = (lane & 16) + 15;
        #pragma unroll
        for (int j = 0; j < 8; ++j) carry[j] = __shfl(c[j], src, 32);
    }
}

// ---------------------------------------------------------------------------
// Kernel 2: column-wise inclusive prefix sum (in place on row-prefixed data).
// out_tile = L(16x16 lower-triangular ones) x T(16x16). A is the constant L;
// B is the data tile. Per-column carry enters via the accumulator, updated by
// shfl from the m=15 row (vgpr 7, lanes 16-31). One wave per 16-col band.
// ---------------------------------------------------------------------------
__global__ __launch_bounds__(32) void col_scan_wmma(
    float* __restrict__ sat, int Hc, int Wc, int colBandsPerImage)
{
    int blk  = blockIdx.x;
    int b    = blk / colBandsPerImage;
    int cb   = blk % colBandsPerImage;
    int col0 = cb * 16;
    int lane = threadIdx.x;
    int half = lane >> 4;
    int m    = lane & 15;
    const size_t base = (size_t)b * Hc * Wc;
    float* colPtr = sat + base + col0 + m;

    // A = L slices (constant): A[m][k] = (k <= m)
    v2f amat[4];
    #pragma unroll
    for (int s = 0; s < 4; ++s) {
        int kb = 4 * s + 2 * half;
        amat[s] = v2f{ (kb     <= m) ? 1.f : 0.f,
                       (kb + 1 <= m) ? 1.f : 0.f };
    }

    float carryc = 0.f;
    int nRowTiles = (Hc + 15) / 16;   // 68 (last tile 8 valid rows)
    for (int ty = 0; ty < nRowTiles; ++ty) {
        int  ry0      = ty * 16;
        bool fullTile = (ry0 + 16 <= Hc);   // wave-uniform

        // load all 4 B slices first; clamped address + 0/1 mask (branch-free)
        v2f bm[4];
        #pragma unroll
        for (int s = 0; s < 4; ++s) {
            int   k0 = ry0 + 4 * s + 2 * half;
            int   k1 = k0 + 1;
            float v0 = colPtr[(size_t)min(k0, Hc - 1) * Wc];
            float v1 = colPtr[(size_t)min(k1, Hc - 1) * Wc];
            bm[s] = v2f{ (k0 < Hc) ? v0 : 0.f,
                         (k1 < Hc) ? v1 : 0.f };
        }
        v8f c;
        #pragma unroll
        for (int j = 0; j < 8; ++j) c[j] = carryc;
        #pragma unroll
        for (int s = 0; s < 4; ++s)
            c = __builtin_amdgcn_wmma_f32_16x16x4_f32(
                    false, amat[s], false, bm[s], (short)0, c, false, false);

        if (fullTile) {
            #pragma unroll
            for (int j = 0; j < 8; ++j)
                colPtr[(size_t)(ry0 + half * 8 + j) * Wc] = c[j];
        } else {
            #pragma unroll
            for (int j = 0; j < 8; ++j) {
                int row = ry0 + half * 8 + j;
                if (row < Hc) colPtr[(size_t)row * Wc] = c[j];
            }
        }
        carryc = __shfl(c[7], 16 + m, 32);  // bottom row (m=15), per column
    }
}

// ---------------------------------------------------------------------------
// Kernel 3: per-keypoint descriptor. One 256-thread block per (b,k); thread p
// handles pair p. Box averages = 4 clamped SAT reads (zero-padding semantics,
// denominator always (2r+1)^2). Block-wide L2 normalization.
// ---------------------------------------------------------------------------
__global__ __launch_bounds__(256) void bad_desc_kernel(
    const float* __restrict__ sat,
    const float* __restrict__ kps,
    const float* __restrict__ ori,
    const float* __restrict__ oy1, const float* __restrict__ ox1,
    const float* __restrict__ oy2, const float* __restrict__ ox2,
    const float* __restrict__ thr, const int* __restrict__ radii,
    float* __restrict__ out, int Hc, int Wc, int Kc, int Pc)
{
    int bk = blockIdx.x;
    int b  = bk / Kc;
    int p  = threadIdx.x;

    float ky = kps[(size_t)bk * 2 + 0];
    float kx = kps[(size_t)bk * 2 + 1];
    float valid = (ky >= 0.f) ? 1.f : 0.f;
    float yc = fminf(fmaxf(ky, 0.f), (float)(Hc - 1));
    float xc = fminf(fmaxf(kx, 0.f), (float)(Wc - 1));
    int iy = (int)rintf(yc);   // rintf = round-half-even, matches jnp.round
    int ix = (int)rintf(xc);
    const size_t base = (size_t)b * Hc * Wc;
    float th = ori[base + (size_t)iy * Wc + ix];
    float st, ct;
    sincosf(th, &st, &ct);

    float a_y1 = oy1[p], a_x1 = ox1[p], a_y2 = oy2[p], a_x2 = ox2[p];
    int   r    = radii[p];

    float rdy1 = a_x1 * st + a_y1 * ct, rdx1 = a_x1 * ct - a_y1 * st;
    float rdy2 = a_x2 * st + a_y2 * ct, rdx2 = a_x2 * ct - a_y2 * st;

    auto S = [&](int yy, int xx) -> float {
        return sat[base + (size_t)yy * Wc + xx];
    };
    auto boxavg = [&](float py, float px) -> float {
        int jy = (int)fminf(fmaxf(rintf(py), 0.f), (float)(Hc - 1));
        int jx = (int)fminf(fmaxf(rintf(px), 0.f), (float)(Wc - 1));
        int y2 = min(jy + r, Hc - 1), x2 = min(jx + r, Wc - 1);
        int y1 = jy - r,              x1 = jx - r;
        float s = S(y2, x2);
        if (y1 > 0)            s -= S(y1 - 1, x2);
        if (x1 > 0)            s -= S(y2, x1 - 1);
        if (y1 > 0 && x1 > 0)  s += S(y1 - 1, x1 - 1);
        float w = 2.f * (float)r + 1.f;
        return s / (w * w);
    };

    float v1 = boxavg(yc + rdy1, xc + rdx1);
    float v2 = boxavg(yc + rdy2, xc + rdx2);
    float d  = v1 - v2 - thr[p];

    // L2 norm over the 256 descriptor elements (8 wave32 waves)
    float sq = d * d;
    #pragma unroll
    for (int off = 16; off > 0; off >>= 1) sq += __shfl_down(sq, off, 32);
    __shared__ float wacc[8];
    if ((p & 31) == 0) wacc[p >> 5] = sq;
    __syncthreads();
    float ss = 0.f;
    #pragma unroll
    for (int w = 0; w < 8; ++w) ss += wacc[w];
    float nrm = fmaxf(sqrtf(ss), 1e-12f);

    out[(size_t)bk * Pc + p] = d / nrm * valid;
}

extern "C" void kernel_launch(void* const* d_in, const int* in_sizes, int n_in,
                              void* d_out, int out_size, void* d_ws, size_t ws_size,
                              hipStream_t stream)
{
    const int B = 2, H = 1080, W = 1920, K = 2048, P = 256;
    const float* image = (const float*)d_in[0];
    const float* kps   = (const float*)d_in[1];
    const float* ori   = (const float*)d_in[2];
    const float* oy1   = (const float*)d_in[3];
    const float* ox1   = (const float*)d_in[4];
    const float* oy2   = (const float*)d_in[5];
    const float* ox2   = (const float*)d_in[6];
    const float* thr   = (const float*)d_in[7];
    const int*   rad   = (const int*)d_in[8];
    float* sat = (float*)d_ws;          // needs B*H*W*4 = 16.6 MB scratch
    float* out = (float*)d_out;

    int bands  = (H + 15) / 16;  // 68 row bands per image
    int cbands = W / 16;         // 120 column bands per image

    row_scan_wmma<<<dim3(B * bands),  dim3(32),  0, stream>>>(image, sat, H, W, bands);
    col_scan_wmma<<<dim3(B * cbands), dim3(32),  0, stream>>>(sat, H, W, cbands);
    bad_desc_kernel<<<dim3(B * K),    dim3(256), 0, stream>>>(
        sat, kps, ori, oy1, ox1, oy2, ox2, thr, rad, out, H, W, K, P);
}